// DFTD3_74294344286992
// MI455X (gfx1250) — compile-verified
//
#include <hip/hip_runtime.h>
#include <hip/hip_bf16.h>

// ---------------- D3 constants ----------------
#define D3_K1   16.0f
#define D3_K3    4.0f
#define D3_A1    0.4f
#define D3_A2    4.8f
#define D3_S6    1.0f
#define D3_S8    2.0f
#define D3_CN2   625.0f    // CN_CUTOFF^2
#define D3_D2    2500.0f   // DISP_CUTOFF^2
#define D3_EPS   1e-20f
#define ZMAX     95
#define MREF     5

typedef __attribute__((ext_vector_type(2))) float v2f;
typedef __attribute__((ext_vector_type(8))) float v8f;

// Hardware approx ops: single TRANS instructions (v_rcp/v_rsq/v_sqrt), ~1 ulp.
__device__ __forceinline__ float fast_rcp(float x)  { return __builtin_amdgcn_rcpf(x); }
__device__ __forceinline__ float fast_rsq(float x)  { return __builtin_amdgcn_rsqf(x); }
__device__ __forceinline__ float fast_sqrt(float x) { return __builtin_amdgcn_sqrtf(x); }

// Wave32 sum-reduction via one v_wmma_f32_16x16x4_f32.
// A holds one partial per lane (other slot 0), B = all-ones, C = 0.
// D[m][n] = rowsumA[m] for every n. Summing a lane's 8 D registers yields
// sum(rows 0-7) in lanes 0-15 and sum(rows 8-15) in lanes 16-31,
// so the two half-sums together are the full wave sum.
// Requires EXEC all-ones: call uniformly from fully-populated waves.
__device__ __forceinline__ float wave_half_sum_wmma(float acc) {
    v2f a; a.x = acc;  a.y = 0.0f;
    v2f b; b.x = 1.0f; b.y = 1.0f;
    v8f c = {};
    v8f d = __builtin_amdgcn_wmma_f32_16x16x4_f32(
        /*neg_a=*/false, a, /*neg_b=*/false, b,
        /*c_mod=*/(short)0, c, /*reuse_a=*/false, /*reuse_b=*/false);
    return ((d[0] + d[1]) + (d[2] + d[3])) + ((d[4] + d[5]) + (d[6] + d[7]));
}

// ---------------- Pass 0: zero cn[] scratch and the output scalar ----------
__global__ __launch_bounds__(256) void d3_init_kernel(float* __restrict__ cn,
                                                      float* __restrict__ out,
                                                      int n_atoms) {
    int i = blockIdx.x * blockDim.x + threadIdx.x;
    if (i < n_atoms) cn[i] = 0.0f;
    if (i == 0) out[0] = 0.0f;
}

// ---------------- Pass 1: coordination numbers ------------------------------
// cn[i] += sigmoid(K1*((rcov_zi + rcov_zj)/r - 1)) for r < CN_CUTOFF.
// rc/r computed as rc * v_rsq(d2+eps); sigmoid via v_exp + v_rcp.
__global__ __launch_bounds__(256) void d3_cn_kernel(
    const float* __restrict__ pos, const int* __restrict__ znum,
    const int* __restrict__ pair_i, const int* __restrict__ pair_j,
    const float* __restrict__ rcov, float* __restrict__ cn, int n_pairs) {
    int e = blockIdx.x * blockDim.x + threadIdx.x;
    if (e >= n_pairs) return;
    int i = pair_i[e];
    int j = pair_j[e];
    float dx = pos[3 * j + 0] - pos[3 * i + 0];
    float dy = pos[3 * j + 1] - pos[3 * i + 1];
    float dz = pos[3 * j + 2] - pos[3 * i + 2];
    float d2 = dx * dx + dy * dy + dz * dz + D3_EPS;
    if (d2 < D3_CN2) {
        float rinv = fast_rsq(d2);
        float rc = rcov[znum[i]] + rcov[znum[j]];
        float ex = __expf(-D3_K1 * (rc * rinv - 1.0f));
        float cf = fast_rcp(1.0f + ex);
        atomicAdd(&cn[i], cf);
    }
}

// ---------------- Pass 2: C6 interpolation + BJ-damped energy ---------------
// Separable weights: w[m1][m2] = wi[m1]*wj[m2]; norm = (sum wi)(sum wj);
// numerator = wi^T * C6(zi,zj) * wj (C6 block = 25 contiguous floats).
// r enters only via r^2 -> use d2 directly; no distance sqrt needed.
__global__ __launch_bounds__(256) void d3_energy_kernel(
    const float* __restrict__ pos, const int* __restrict__ znum,
    const int* __restrict__ pair_i, const int* __restrict__ pair_j,
    const float* __restrict__ r4r2, const float* __restrict__ c6_tab,
    const float* __restrict__ cn_ref, const float* __restrict__ cn,
    float* __restrict__ out, int n_pairs) {
    float acc = 0.0f;
    int stride = gridDim.x * blockDim.x;
    for (int e = blockIdx.x * blockDim.x + threadIdx.x; e < n_pairs; e += stride) {
        int i  = pair_i[e];
        int j  = pair_j[e];
        int zi = znum[i];
        int zj = znum[j];

        float dx = pos[3 * j + 0] - pos[3 * i + 0];
        float dy = pos[3 * j + 1] - pos[3 * i + 1];
        float dz = pos[3 * j + 2] - pos[3 * i + 2];
        float r2 = dx * dx + dy * dy + dz * dz + D3_EPS;

        float cni = cn[i];
        float cnj = cn[j];

        const float* __restrict__ ri = cn_ref + zi * MREF;
        const float* __restrict__ rj = cn_ref + zj * MREF;
        float wi[MREF], wj[MREF];
        float swi = 0.0f, swj = 0.0f;
#pragma unroll
        for (int m = 0; m < MREF; ++m) {
            float rv = ri[m];
            float d0 = cni - rv;
            float w0 = (rv >= 0.0f) ? __expf(-D3_K3 * d0 * d0) : 0.0f;
            wi[m] = w0; swi += w0;
            float rw = rj[m];
            float d1 = cnj - rw;
            float w1 = (rw >= 0.0f) ? __expf(-D3_K3 * d1 * d1) : 0.0f;
            wj[m] = w1; swj += w1;
        }

        const float* __restrict__ c6b = c6_tab + ((size_t)(zi * ZMAX + zj)) * (MREF * MREF);
        float numer = 0.0f;
#pragma unroll
        for (int m1 = 0; m1 < MREF; ++m1) {
            float u = 0.0f;
#pragma unroll
            for (int m2 = 0; m2 < MREF; ++m2)
                u += c6b[m1 * MREF + m2] * wj[m2];
            numer += wi[m1] * u;
        }
        float norm = swi * swj;
        float c6p  = numer * fast_rcp(norm + D3_EPS);

        float qq  = 3.0f * r4r2[zi] * r4r2[zj];
        float c8p = c6p * qq;
        float r0  = D3_A1 * fast_sqrt(qq) + D3_A2;
        float r6  = r2 * r2 * r2,  r8 = r6 * r2;
        float q2  = r0 * r0, q6 = q2 * q2 * q2, q8 = q6 * q2;
        float ep  = -(D3_S6 * c6p * fast_rcp(r6 + q6) +
                      D3_S8 * c8p * fast_rcp(r8 + q8));
        if (r2 < D3_D2) acc += 0.5f * ep;
    }

    // Uniform control flow: every lane of every wave executes the WMMA.
    float half_sum = wave_half_sum_wmma(acc);
    // Lanes 0 and 16 hold the two complementary half-wave sums; the backend's
    // atomic optimizer folds the active lanes into one SALU sum + one atomic.
    if ((threadIdx.x & 15u) == 0u) atomicAdd(out, half_sum);
}

extern "C" void kernel_launch(void* const* d_in, const int* in_sizes, int n_in,
                              void* d_out, int out_size, void* d_ws, size_t ws_size,
                              hipStream_t stream) {
    const float* pos    = (const float*)d_in[0];   // [N,3]
    const int*   znum   = (const int*)  d_in[1];   // [N]
    const int*   pair_i = (const int*)  d_in[2];   // [E]
    const int*   pair_j = (const int*)  d_in[3];   // [E]
    const float* rcov   = (const float*)d_in[4];   // [95]
    const float* r4r2   = (const float*)d_in[5];   // [95]
    const float* c6tab  = (const float*)d_in[6];   // [95,95,5,5]
    const float* cnref  = (const float*)d_in[7];   // [95,5]
    float*       out    = (float*)d_out;           // scalar
    float*       cn     = (float*)d_ws;            // [N] scratch

    int n_atoms = in_sizes[1];
    int n_pairs = in_sizes[2];

    int initBlocks = (n_atoms + 255) / 256;
    d3_init_kernel<<<initBlocks, 256, 0, stream>>>(cn, out, n_atoms);

    int cnBlocks = (n_pairs + 255) / 256;
    d3_cn_kernel<<<cnBlocks, 256, 0, stream>>>(pos, znum, pair_i, pair_j, rcov, cn, n_pairs);

    int eBlocks = 2048;  // grid-stride; ~4 pairs/thread, few same-address atomics
    d3_energy_kernel<<<eBlocks, 256, 0, stream>>>(pos, znum, pair_i, pair_j,
                                                  r4r2, c6tab, cnref, cn,
                                                  out, n_pairs);
}